// GlobalPointer_47029891891277
// MI455X (gfx1250) — compile-verified
//
#include <hip/hip_runtime.h>

#define HEADS 12
#define HEAD_SIZE 64
#define S_LEN 512
#define D_MODEL 1024
#define NQK (HEADS * HEAD_SIZE * 2) /* 1536 */
#define BATCH 16
#define NEGV 1000000000000.0f

typedef __attribute__((ext_vector_type(2))) float v2f;
typedef __attribute__((ext_vector_type(8))) float v8f;

// ---------------------------------------------------------------------------
// CDNA5 async copy global -> LDS (ASYNCcnt-tracked, bypasses VGPRs).
// Builtin signature (from hipcc diagnostic): pointee type is a 16B int vector.
// Falls back to a VGPR round-trip if the builtin is unavailable.
// ---------------------------------------------------------------------------
#if __has_builtin(__builtin_amdgcn_global_load_async_to_lds_b128) && \
    __has_builtin(__builtin_amdgcn_s_wait_asynccnt)
#define USE_ASYNC 1
#endif

typedef int vi4 __attribute__((vector_size(16)));
typedef __attribute__((address_space(1))) vi4 gvi4;   // global
typedef __attribute__((address_space(3))) vi4 lvi4;   // LDS

__device__ __forceinline__ void cp16(float* l, const float* g) {
#ifdef USE_ASYNC
    // Generic LDS pointers carry the LDS byte offset in their low 32 bits.
    __builtin_amdgcn_global_load_async_to_lds_b128(
        (gvi4*)(unsigned long long)g,
        (lvi4*)(unsigned int)(unsigned long long)l,
        0, 0);
#else
    *(float4*)l = *(const float4*)g;
#endif
}

#ifdef USE_ASYNC
#define WAIT_ASYNC(n) __builtin_amdgcn_s_wait_asynccnt(n)
#else
#define WAIT_ASYNC(n) ((void)0)
#endif

// ---------------------------------------------------------------------------
// Kernel 1: fused QK projection + bias + interleaved RoPE
//   C[M=8192][N=1536] = X @ W^T + b ; RoPE ; scatter -> qw/kw (B,H,S,64)
// 64x64 tile, K in chunks of 32, double-buffered async LDS staging.
// 128 threads = 4 waves; wave w owns M strip w*16..w*16+15, all 64 N cols.
// ---------------------------------------------------------------------------
#define K1_KC    32
#define K1_LDA   36  // padded row stride: 144B rows -> 16B aligned, conflict-free b64
#define K1_NCHNK (D_MODEL / K1_KC)

__global__ __launch_bounds__(128) void qk_rope_kernel(
    const float* __restrict__ X,     // (8192, 1024)
    const float* __restrict__ W,     // (1536, 1024) row-major
    const float* __restrict__ bias,  // (1536)
    float* __restrict__ qw,          // (16,12,512,64)
    float* __restrict__ kw)          // (16,12,512,64)
{
    __shared__ float As[2][64 * K1_LDA];
    __shared__ float Bs[2][64 * K1_LDA];

    const int tid  = threadIdx.x;
    const int wave = tid >> 5;
    const int lane = tid & 31;
    const int l16  = lane & 15;
    const int lhi  = lane >> 4;        // 0 or 1

    const int m_base = blockIdx.y * 64;
    const int n_base = blockIdx.x * 64;

    v8f acc[4] = {};                   // 4 N-subtiles of 16x16 f32

    // ---- prologue: stage chunk 0 into buffer 0 (8 async ops / thread) ----
    #pragma unroll
    for (int i = tid; i < 64 * (K1_KC / 4); i += 128) {
        const int r  = i >> 3;
        const int c4 = i & 7;
        cp16(&As[0][r * K1_LDA + c4 * 4], X + (size_t)(m_base + r) * D_MODEL + c4 * 4);
        cp16(&Bs[0][r * K1_LDA + c4 * 4], W + (size_t)(n_base + r) * D_MODEL + c4 * 4);
    }

    for (int c = 0; c < K1_NCHNK; ++c) {
        const int cur = c & 1;
        if (c + 1 < K1_NCHNK) {
            // issue next chunk into the other buffer before computing
            const int k0 = (c + 1) * K1_KC;
            #pragma unroll
            for (int i = tid; i < 64 * (K1_KC / 4); i += 128) {
                const int r  = i >> 3;
                const int c4 = i & 7;
                cp16(&As[cur ^ 1][r * K1_LDA + c4 * 4],
                     X + (size_t)(m_base + r) * D_MODEL + k0 + c4 * 4);
                cp16(&Bs[cur ^ 1][r * K1_LDA + c4 * 4],
                     W + (size_t)(n_base + r) * D_MODEL + k0 + c4 * 4);
            }
            WAIT_ASYNC(8);   // in-order completion: <=8 outstanding == chunk c done
        } else {
            WAIT_ASYNC(0);
        }
        __syncthreads();

        const float* Ac = As[cur];
        const float* Bc = Bs[cur];
        #pragma unroll
        for (int kk = 0; kk < K1_KC; kk += 4) {
            const int ko = kk + lhi * 2;                 // A: lane L -> K=(L/16)*2+{0,1}
            v2f a = *(const v2f*)(Ac + (wave * 16 + l16) * K1_LDA + ko);
            #pragma unroll
            for (int ns = 0; ns < 4; ++ns) {
                v2f bf = *(const v2f*)(Bc + (ns * 16 + l16) * K1_LDA + ko);
                acc[ns] = __builtin_amdgcn_wmma_f32_16x16x4_f32(
                    false, a, false, bf, (short)0, acc[ns], false, false);
            }
        }
        __syncthreads();   // all waves done with buf `cur` before it is re-filled
    }

    // ---- epilogue: bias + interleaved RoPE + scatter into q / k buffers ----
    const int  h    = n_base / (2 * HEAD_SIZE);
    const bool is_k = (n_base & (2 * HEAD_SIZE - 1)) >= HEAD_SIZE;
    float* __restrict__ dst = is_k ? kw : qw;

    #pragma unroll
    for (int ns = 0; ns < 4; ++ns) {
        const int   n    = n_base + ns * 16 + l16;       // C/D: col = lane%16
        const int   d    = n & (HEAD_SIZE - 1);
        const float bval = bias[n];
        const int   j    = d >> 1;
        const float inv  = __expf(-0.28782313662f * (float)j); // 10000^(-j/32)
        #pragma unroll
        for (int v = 0; v < 8; ++v) {
            const int m     = m_base + wave * 16 + lhi * 8 + v;  // row = (L/16)*8+v
            const int s     = m & (S_LEN - 1);
            const int batch = m >> 9;
            float val = acc[ns][v] + bval;
            // rotate_half partner x[d^1] sits in the adjacent lane
            float partner = __shfl_xor(val, 1, 32);
            float ang = (float)s * inv;
            float sn, cs;
            __sincosf(ang, &sn, &cs);
            float rot = (d & 1) ? partner : -partner;
            float r = val * cs + rot * sn;
            dst[(((size_t)batch * HEADS + h) * S_LEN + s) * HEAD_SIZE + d] = r;
        }
    }
}

// ---------------------------------------------------------------------------
// Kernel 2: attention logits per (b,h): 64x64 tile of Q @ K^T, K-dim = 64
// staged once in LDS (async); fused padding mask, tril mask, 1/8 scale.
// ---------------------------------------------------------------------------
#define K2_LD 68   // 272B rows: 16B-aligned b128 stores, conflict-free b64 reads

__global__ __launch_bounds__(128) void attn_logits_kernel(
    const float* __restrict__ qw,    // (16,12,512,64)
    const float* __restrict__ kw,    // (16,12,512,64)
    const float* __restrict__ mask,  // (16,512)
    float* __restrict__ out)         // (16,12,512,512)
{
    __shared__ float Qs[64 * K2_LD];
    __shared__ float Ks[64 * K2_LD];

    const int tid  = threadIdx.x;
    const int wave = tid >> 5;
    const int lane = tid & 31;
    const int l16  = lane & 15;
    const int lhi  = lane >> 4;

    const int bh     = blockIdx.z;          // b*12 + h
    const int b      = bh / HEADS;
    const int m_tile = blockIdx.y * 64;
    const int n_tile = blockIdx.x * 64;

    const float* qbase = qw + ((size_t)bh * S_LEN + m_tile) * HEAD_SIZE;
    const float* kbase = kw + ((size_t)bh * S_LEN + n_tile) * HEAD_SIZE;

    #pragma unroll
    for (int i = tid; i < 64 * (HEAD_SIZE / 4); i += 128) {
        const int r  = i >> 4;
        const int c4 = i & 15;
        cp16(Qs + r * K2_LD + c4 * 4, qbase + (size_t)r * HEAD_SIZE + c4 * 4);
        cp16(Ks + r * K2_LD + c4 * 4, kbase + (size_t)r * HEAD_SIZE + c4 * 4);
    }
    WAIT_ASYNC(0);
    __syncthreads();

    v8f acc[4] = {};
    #pragma unroll
    for (int k0 = 0; k0 < HEAD_SIZE; k0 += 4) {
        const int ko = k0 + lhi * 2;
        v2f a = *(const v2f*)(Qs + (wave * 16 + l16) * K2_LD + ko);
        #pragma unroll
        for (int ns = 0; ns < 4; ++ns) {
            v2f bf = *(const v2f*)(Ks + (ns * 16 + l16) * K2_LD + ko);
            acc[ns] = __builtin_amdgcn_wmma_f32_16x16x4_f32(
                false, a, false, bf, (short)0, acc[ns], false, false);
        }
    }

    #pragma unroll
    for (int ns = 0; ns < 4; ++ns) {
        const int   n   = n_tile + ns * 16 + l16;
        const float pad = mask[(size_t)b * S_LEN + n];
        const float neg_term = (1.0f - pad) * NEGV;
        #pragma unroll
        for (int v = 0; v < 8; ++v) {
            const int m = m_tile + wave * 16 + lhi * 8 + v;
            float val = acc[ns][v] * pad - neg_term;
            if (n < m) val -= NEGV;                     // tril(ones,-1) mask
            val *= 0.125f;                              // 1/sqrt(64)
            out[((size_t)bh * S_LEN + m) * S_LEN + n] = val;
        }
    }
}

// ---------------------------------------------------------------------------
extern "C" void kernel_launch(void* const* d_in, const int* in_sizes, int n_in,
                              void* d_out, int out_size, void* d_ws, size_t ws_size,
                              hipStream_t stream) {
    const float* X    = (const float*)d_in[0];   // inputs (16,512,1024)
    const float* mask = (const float*)d_in[1];   // attention_mask (16,512)
    const float* W    = (const float*)d_in[2];   // W (1536,1024)
    const float* bias = (const float*)d_in[3];   // b (1536)
    float* out = (float*)d_out;

    float* qw = (float*)d_ws;                                   // 25.2 MB
    float* kw = qw + (size_t)BATCH * HEADS * S_LEN * HEAD_SIZE; // 25.2 MB

    dim3 g1(NQK / 64, (BATCH * S_LEN) / 64);    // 24 x 128
    qk_rope_kernel<<<g1, 128, 0, stream>>>(X, W, bias, qw, kw);

    dim3 g2(S_LEN / 64, S_LEN / 64, BATCH * HEADS);   // 8 x 8 x 192
    attn_logits_kernel<<<g2, 128, 0, stream>>>(qw, kw, mask, out);
}